// LocallyConnected2D_fast_41867341201420
// MI455X (gfx1250) — compile-verified
//
#include <hip/hip_runtime.h>

typedef __attribute__((ext_vector_type(2))) float v2f;
typedef __attribute__((ext_vector_type(4))) float v4f;
typedef __attribute__((ext_vector_type(8))) float v8f;

// LDS layout (dwords). Row stride 18 (16 w + 2 pad) => conflict-free b64 frag
// reads across lid; dr-block pad of 32 dwords shifts banks of lanes 16-31 by
// 32 so both wave halves hit disjoint banks.
#define XS_DRBLK 608     // 32 rows * 18 + 32
#define XS_CBLK  1216    // 2 * XS_DRBLK
#define WS_BASE  9728    // 8 * XS_CBLK
#define WS_DRBLK 1184    // 64 rows * 18 + 32
#define WS_CBLK  2368
#define LDS_DWORDS 28672 // WS_BASE + 8*WS_CBLK = 112 KB
#define OS_BSTRIDE 580   // 64*9 + 4 (epilogue staging, overlays XS/WS)

__global__ __launch_bounds__(256, 1) void lc2d_wmma(
    const float* __restrict__ xp, const float* __restrict__ wp,
    const float* __restrict__ bp, float* __restrict__ out) {
  extern __shared__ float lds[];
  const int tid  = threadIdx.x;
  const int lane = tid & 31;
  const int lid  = lane & 15;
  const int half = lane >> 4;
  const int p    = tid >> 5;          // wave id = pixel within oc-block (0..7)

  const int bid = blockIdx.x;         // 512 blocks
  const int orr = bid >> 3;           // output row (0..63)
  const int ocb = bid & 7;            // oc block
  const int oc0 = ocb << 3;           // first oc of block
  const int oc  = oc0 + p;            // this wave's oc
  const int h0  = orr << 1;           // first input row
  const int w0  = oc0 << 1;           // first input col (mult of 16 -> 64B aligned)
  const int twop = 2 * p;

  // Bias: reference raw-reshapes b(OR,OC,F) -> (1,F,OR,OC):
  // bias(f,or,oc) = b_flat[f*4096 + or*64 + oc]
  float bv[4];
#pragma unroll
  for (int n0 = 0; n0 < 4; ++n0)
    bv[n0] = bp[(n0 * 16 + lid) * 4096 + orr * 64 + oc];

  v8f acc[2][4] = {};

  const int w2 = (tid & 7) * 2;       // starting w within 16-float row (8B units)

  for (int cc = 0; cc < 4; ++cc) {
    const int cbase = cc * 8;
    __syncthreads();

    // ---- cooperative global -> LDS, 64B-line coalesced (8 lanes per row) ----
    {
      int r = tid >> 3;               // 512 x-rows: (c,dr,b), b fastest
      for (int it = 0; it < 16; ++it, r += 32) {
        const int b = r & 31, dr = (r >> 5) & 1, c = r >> 6;
        const float* gp = xp + (((b * 32 + cbase + c) * 128 + (h0 + dr)) * 128 + w0 + w2);
        *(v2f*)(lds + c * XS_CBLK + dr * XS_DRBLK + b * 18 + w2) = *(const v2f*)gp;
      }
      r = tid >> 3;                   // 1024 W-rows: (c,dr,f), f fastest
      for (int it = 0; it < 32; ++it, r += 32) {
        const int f = r & 63, dr = (r >> 6) & 1, c = r >> 7;
        const float* gp = wp + (((f * 32 + cbase + c) * 128 + (h0 + dr)) * 128 + w0 + w2);
        *(v2f*)(lds + WS_BASE + c * WS_CBLK + dr * WS_DRBLK + f * 18 + w2) = *(const v2f*)gp;
      }
    }

    // ---- prefetch next chunk's cachelines (overlaps the compute below) ----
    if (cc < 3) {
      const int nb = cbase + 8;
      int line = tid;                 // 1536 lines total (512 x + 1024 W)
      for (int q = 0; q < 6; ++q, line += 256) {
        const float* gp;
        if (line < 512) {
          const int b = line & 31, dr = (line >> 5) & 1, c = line >> 6;
          gp = xp + (((b * 32 + nb + c) * 128 + (h0 + dr)) * 128 + w0);
        } else {
          const int l2 = line - 512;
          const int f = l2 & 63, dr = (l2 >> 6) & 1, c = l2 >> 7;
          gp = wp + (((f * 32 + nb + c) * 128 + (h0 + dr)) * 128 + w0);
        }
        __builtin_prefetch(gp, 0, 1);
      }
    }

    __syncthreads();

    // ---- compute: 8 K-steps of V_WMMA_F32_16X16X4_F32 per chunk ----
    // Fragment element K = 2*half + j  <->  (dr = half, dc = j): one v2f
    // (ds_load_b64) of two consecutive w floats per fragment.
    const float* axb = lds + half * XS_DRBLK + lid * 18 + twop;
    const float* bxb = lds + WS_BASE + half * WS_DRBLK + lid * 18 + twop;
#pragma unroll
    for (int kk = 0; kk < 8; ++kk) {
      const float* ax = axb + kk * XS_CBLK;
      const float* bx = bxb + kk * WS_CBLK;
      v2f a0 = *(const v2f*)(ax);
      v2f a1 = *(const v2f*)(ax + 16 * 18);
      v2f b0 = *(const v2f*)(bx);
      v2f b1 = *(const v2f*)(bx + 16 * 18);
      v2f b2 = *(const v2f*)(bx + 32 * 18);
      v2f b3 = *(const v2f*)(bx + 48 * 18);
      acc[0][0] = __builtin_amdgcn_wmma_f32_16x16x4_f32(false, a0, false, b0, (short)0, acc[0][0], false, false);
      acc[0][1] = __builtin_amdgcn_wmma_f32_16x16x4_f32(false, a0, false, b1, (short)0, acc[0][1], false, false);
      acc[0][2] = __builtin_amdgcn_wmma_f32_16x16x4_f32(false, a0, false, b2, (short)0, acc[0][2], false, false);
      acc[0][3] = __builtin_amdgcn_wmma_f32_16x16x4_f32(false, a0, false, b3, (short)0, acc[0][3], false, false);
      acc[1][0] = __builtin_amdgcn_wmma_f32_16x16x4_f32(false, a1, false, b0, (short)0, acc[1][0], false, false);
      acc[1][1] = __builtin_amdgcn_wmma_f32_16x16x4_f32(false, a1, false, b1, (short)0, acc[1][1], false, false);
      acc[1][2] = __builtin_amdgcn_wmma_f32_16x16x4_f32(false, a1, false, b2, (short)0, acc[1][2], false, false);
      acc[1][3] = __builtin_amdgcn_wmma_f32_16x16x4_f32(false, a1, false, b3, (short)0, acc[1][3], false, false);
    }
  }

  // ---- epilogue: bias + relu, stage through LDS, coalesced 32B stores ----
  __syncthreads();                    // all waves done reading XS/WS
#pragma unroll
  for (int m0 = 0; m0 < 2; ++m0) {
#pragma unroll
    for (int n0 = 0; n0 < 4; ++n0) {
      const int f    = n0 * 16 + lid; // D-matrix: N = lid
      const int brow = m0 * 16 + half * 8;
      float* osp = lds + brow * OS_BSTRIDE + f * 9 + p;
      const float bb = bv[n0];
#pragma unroll
      for (int v = 0; v < 8; ++v) {   // D-matrix: M = v + 8*half
        float val = acc[m0][n0][v] + bb;
        osp[v * OS_BSTRIDE] = fmaxf(val, 0.0f);
      }
    }
  }
  __syncthreads();

  int pair = tid;                     // 2048 (b,f) pairs, 8 per thread
  for (int it = 0; it < 8; ++it, pair += 256) {
    const int b = pair >> 6, f = pair & 63;
    const float* sp = lds + b * OS_BSTRIDE + f * 9;
    v4f lo = {sp[0], sp[1], sp[2], sp[3]};
    v4f hi = {sp[4], sp[5], sp[6], sp[7]};
    float* op = out + (((b * 64 + f) * 64 + orr) * 64 + oc0);
    *(v4f*)(op) = lo;
    *(v4f*)(op + 4) = hi;
  }
}

extern "C" void kernel_launch(void* const* d_in, const int* in_sizes, int n_in,
                              void* d_out, int out_size, void* d_ws, size_t ws_size,
                              hipStream_t stream) {
  const float* x  = (const float*)d_in[0];
  const float* W  = (const float*)d_in[1];
  const float* b  = (const float*)d_in[2];
  float* out = (float*)d_out;

  const size_t lds_bytes = (size_t)LDS_DWORDS * sizeof(float); // 112 KB
  // Allow >default dynamic LDS (gfx1250 WGP supports up to 320 KB).
  (void)hipFuncSetAttribute((const void*)lc2d_wmma,
                            hipFuncAttributeMaxDynamicSharedMemorySize,
                            (int)lds_bytes);
  lc2d_wmma<<<dim3(512), dim3(256), lds_bytes, stream>>>(x, W, b, out);
}